// RecurrentGCN_28329604284505
// MI455X (gfx1250) — compile-verified
//
#include <hip/hip_runtime.h>

typedef __attribute__((ext_vector_type(2))) float v2f;
typedef __attribute__((ext_vector_type(8))) float v8f;

#define NNODES 50000
#define NEDGES 800000
#define DIM 128
#define GEMM_N 128

// ---------------------------------------------------------------------------
// WMMA fp32 GEMM: C[M x 128] = A[M x K] @ W[K x 128] (+ bias) (+= C if add)
// Block = 256 threads = 8 waves. Each block owns 32 rows of A (staged in LDS);
// wave w owns columns [16w, 16w+16) and computes TWO 16x16 output tiles
// (rows [m0, m0+16) and [m0+16, m0+32)) so each B fragment load feeds two
// v_wmma_f32_16x16x4_f32 ops -> half the W traffic per FLOP vs 1-tile waves.
//
// Fragment layouts per CDNA5 ISA 7.12.2 (wave32):
//   A 16x4 f32 : lanes 0-15 hold M=lane, VGPR0=K0,VGPR1=K1; lanes 16-31 K2,K3
//   B 4x16 f32 : lanes 0-15 hold N=lane, rows K0,K1; lanes 16-31 rows K2,K3
//   C/D 16x16  : VGPR r -> row r (lanes 0-15) / row r+8 (lanes 16-31), col=lane%16
// ---------------------------------------------------------------------------
__launch_bounds__(256)
__global__ void wmma_gemm_f32(const float* __restrict__ A,
                              const float* __restrict__ W,
                              float* __restrict__ C,
                              const float* __restrict__ bias,
                              int M, int K, int add) {
  __shared__ float As[32 * (DIM + 4)];
  const int tid  = threadIdx.x;
  const int wave = tid >> 5;
  const int lane = tid & 31;
  const int m0   = blockIdx.x << 5;   // 32 rows per block
  const int lds  = K + 4;             // padded row stride (132 floats @ K=128)

  // Stage the 32 x K A-tile into LDS (float4; row index clamped at the edge,
  // duplicated loads are harmless and stores are guarded below).
  for (int i = tid; i < (32 * K) / 4; i += 256) {
    const int r  = i / (K / 4);
    const int c  = (i % (K / 4)) * 4;
    const int gr = min(m0 + r, M - 1);
    const float4 v = *(const float4*)(A + (size_t)gr * K + c);
    *(float4*)(&As[r * lds + c]) = v;
  }
  __syncthreads();

  const int n0   = wave << 4;
  const int lmod = lane & 15;
  const int kh   = (lane >> 4) << 1;  // 0 for lanes 0-15, 2 for lanes 16-31

  v8f acc[2];
  acc[0] = (v8f){};
  acc[1] = (v8f){};
  for (int k0 = 0; k0 < K; k0 += 4) {
    const int kk = k0 + kh;
    v2f a0, a1, b;
    b[0]  = W[(size_t)kk * GEMM_N + n0 + lmod];
    b[1]  = W[(size_t)(kk + 1) * GEMM_N + n0 + lmod];
    a0[0] = As[lmod * lds + kk];
    a0[1] = As[lmod * lds + kk + 1];
    a1[0] = As[(lmod + 16) * lds + kk];
    a1[1] = As[(lmod + 16) * lds + kk + 1];
    acc[0] = __builtin_amdgcn_wmma_f32_16x16x4_f32(
        false, a0, false, b, (short)0, acc[0], false, false);
    acc[1] = __builtin_amdgcn_wmma_f32_16x16x4_f32(
        false, a1, false, b, (short)0, acc[1], false, false);
  }

  const int col  = n0 + lmod;
  const float bv = bias ? bias[col] : 0.0f;
#pragma unroll
  for (int mt = 0; mt < 2; ++mt) {
    const int rbase = m0 + (mt << 4) + ((lane >> 4) << 3);
#pragma unroll
    for (int r = 0; r < 8; ++r) {
      const int row = rbase + r;
      if (row < M) {
        const size_t idx = (size_t)row * GEMM_N + col;
        float v = acc[mt][r] + bv;
        if (add) v += C[idx];
        C[idx] = v;
      }
    }
  }
}

// ---------------------------------------------------------------------------
// Graph normalization: deg[i] = 1 (self loop) + #incoming edges; dinv = rsqrt.
// ---------------------------------------------------------------------------
__global__ void deg_init(float* deg, int n) {
  const int i = blockIdx.x * blockDim.x + threadIdx.x;
  if (i < n) deg[i] = 1.0f;
}

__global__ void deg_count(const int* __restrict__ dst, float* deg, int e) {
  const int i = blockIdx.x * blockDim.x + threadIdx.x;
  if (i < e) atomicAdd(&deg[dst[i]], 1.0f);
}

__global__ void dinv_finalize(float* deg, int n) {
  const int i = blockIdx.x * blockDim.x + threadIdx.x;
  if (i < n) {
    const float d = deg[i];
    deg[i] = d > 0.0f ? rsqrtf(d) : 0.0f;
  }
}

// ---------------------------------------------------------------------------
// agg_g = dinv^2 * xw_g (self loop) + bias; edges then atomically accumulate.
// ---------------------------------------------------------------------------
__global__ void agg_init(const float* __restrict__ dinv,
                         const float* __restrict__ xwz, const float* __restrict__ xwr,
                         const float* __restrict__ xwh,
                         const float* __restrict__ bcz, const float* __restrict__ bcr,
                         const float* __restrict__ bch,
                         float* __restrict__ aggz, float* __restrict__ aggr,
                         float* __restrict__ aggh) {
  const size_t i = (size_t)blockIdx.x * 256 + threadIdx.x;  // grid is exact
  const int node = (int)(i >> 7);
  const int f    = (int)(i & 127);
  float c = dinv[node];
  c *= c;
  aggz[i] = c * xwz[i] + bcz[f];
  aggr[i] = c * xwr[i] + bcr[f];
  aggh[i] = c * xwh[i] + bch[f];
}

// One wave per edge; lane l covers features l, l+32, l+64, l+96 for all 3
// convs. Loads are 128B-coalesced per wave; adds are global_atomic_add_f32,
// served from L2 (all six feature arrays fit in the 192MB L2). The edge-index
// streams are sequential, so prefetch a few KB ahead (global_prefetch_b8;
// speculative prefetch past the end of the array is silently dropped).
__global__ void edge_scatter(const int* __restrict__ src, const int* __restrict__ dst,
                             const float* __restrict__ dinv,
                             const float* __restrict__ xwz, const float* __restrict__ xwr,
                             const float* __restrict__ xwh,
                             float* __restrict__ aggz, float* __restrict__ aggr,
                             float* __restrict__ aggh, int e) {
  const int wid  = (int)(((size_t)blockIdx.x * blockDim.x + threadIdx.x) >> 5);
  const int lane = threadIdx.x & 31;
  if (wid >= e) return;
  __builtin_prefetch(src + wid + 4096, 0, 1);
  __builtin_prefetch(dst + wid + 4096, 0, 1);
  const int s = src[wid];
  const int d = dst[wid];
  const float c = dinv[s] * dinv[d];
  const size_t sb = (size_t)s * DIM;
  const size_t db = (size_t)d * DIM;
#pragma unroll
  for (int k = 0; k < 4; ++k) {
    const int f = lane + (k << 5);
    atomicAdd(&aggz[db + f], c * xwz[sb + f]);
    atomicAdd(&aggr[db + f], c * xwr[sb + f]);
    atomicAdd(&aggh[db + f], c * xwh[sb + f]);
  }
}

// ---------------------------------------------------------------------------
// Elementwise gate math.
// ---------------------------------------------------------------------------
__global__ void gates_zr(float* __restrict__ Zp, float* __restrict__ Rp,
                         const float* __restrict__ h, float* __restrict__ hr) {
  const size_t i = (size_t)blockIdx.x * 256 + threadIdx.x;  // grid is exact
  const float z = 1.0f / (1.0f + expf(-Zp[i]));
  const float r = 1.0f / (1.0f + expf(-Rp[i]));
  Zp[i] = z;
  Rp[i] = r;
  hr[i] = h[i] * r;
}

__global__ void gru_combine(const float* __restrict__ Z, const float* __restrict__ Hp,
                            const float* __restrict__ h,
                            float* __restrict__ h0_out, float* __restrict__ relu_h0) {
  const size_t i = (size_t)blockIdx.x * 256 + threadIdx.x;  // grid is exact
  const float ht = tanhf(Hp[i]);
  const float z  = Z[i];
  const float v  = z * h[i] + (1.0f - z) * ht;
  h0_out[i]  = v;
  relu_h0[i] = v > 0.0f ? v : 0.0f;
}

// ---------------------------------------------------------------------------
// Orchestration. Inputs (setup_inputs order):
//  0 node_feat  1 src  2 dst  3 h
//  4 Wc_z 5 bc_z 6 Wl_z 7 bl_z   8 Wc_r 9 bc_r 10 Wl_r 11 bl_r
// 12 Wc_hh 13 bc_hh 14 Wl_hh 15 bl_hh   16 W_lin 17 b_lin
// Output: [z_out (50000x128), h_0 (50000x128)] concatenated.
// ---------------------------------------------------------------------------
extern "C" void kernel_launch(void* const* d_in, const int* in_sizes, int n_in,
                              void* d_out, int out_size, void* d_ws, size_t ws_size,
                              hipStream_t stream) {
  const float* node_feat = (const float*)d_in[0];
  const int*   src       = (const int*)d_in[1];
  const int*   dst       = (const int*)d_in[2];
  const float* h         = (const float*)d_in[3];
  const float* Wc_z = (const float*)d_in[4];
  const float* bc_z = (const float*)d_in[5];
  const float* Wl_z = (const float*)d_in[6];
  const float* bl_z = (const float*)d_in[7];
  const float* Wc_r = (const float*)d_in[8];
  const float* bc_r = (const float*)d_in[9];
  const float* Wl_r = (const float*)d_in[10];
  const float* bl_r = (const float*)d_in[11];
  const float* Wc_h = (const float*)d_in[12];
  const float* bc_h = (const float*)d_in[13];
  const float* Wl_h = (const float*)d_in[14];
  const float* bl_h = (const float*)d_in[15];
  const float* W_lin = (const float*)d_in[16];
  const float* b_lin = (const float*)d_in[17];

  const size_t R = (size_t)NNODES * DIM;  // 6.4M floats per feature region
  float* ws   = (float*)d_ws;
  float* dinv = ws;              // degrees -> dinv (50048 floats, padded)
  float* bufA = ws + 50048;      // xw_z  -> Zpre -> Z
  float* bufB = bufA + R;        // xw_r  -> Rpre -> R
  float* bufC = bufB + R;        // xw_hh -> h*R
  float* bufD = bufC + R;        // agg_z -> Hpre
  float* bufE = bufD + R;        // agg_r -> relu(h0)
  float* bufF = bufE + R;        // agg_hh

  float* out_z  = (float*)d_out;
  float* out_h0 = out_z + R;

  const dim3 blk(256);
  const dim3 nodeGrid((NNODES + 255) / 256);
  const dim3 edgeGrid((NEDGES + 255) / 256);
  const dim3 gemmGrid((NNODES + 31) / 32);   // 1563 blocks, 32 rows each
  const dim3 featGrid((unsigned)(R / 256));  // 25000, exact
  const dim3 waveGrid(NEDGES / 8);           // 8 waves/block, exact

  // 1) symmetric-norm coefficients
  deg_init<<<nodeGrid, blk, 0, stream>>>(dinv, NNODES);
  deg_count<<<edgeGrid, blk, 0, stream>>>(dst, dinv, NEDGES);
  dinv_finalize<<<nodeGrid, blk, 0, stream>>>(dinv, NNODES);

  // 2) xw_g = node_feat @ Wc_g
  wmma_gemm_f32<<<gemmGrid, blk, 0, stream>>>(node_feat, Wc_z, bufA, nullptr, NNODES, DIM, 0);
  wmma_gemm_f32<<<gemmGrid, blk, 0, stream>>>(node_feat, Wc_r, bufB, nullptr, NNODES, DIM, 0);
  wmma_gemm_f32<<<gemmGrid, blk, 0, stream>>>(node_feat, Wc_h, bufC, nullptr, NNODES, DIM, 0);

  // 3) GCN aggregation (self loop + bias init, then edge atomics)
  agg_init<<<featGrid, blk, 0, stream>>>(dinv, bufA, bufB, bufC,
                                         bc_z, bc_r, bc_h, bufD, bufE, bufF);
  edge_scatter<<<waveGrid, blk, 0, stream>>>(src, dst, dinv, bufA, bufB, bufC,
                                             bufD, bufE, bufF, NEDGES);

  // 4) Zpre = agg_z @ Wl_z[:128] + h @ Wl_z[128:] + bl_z   (same for R)
  wmma_gemm_f32<<<gemmGrid, blk, 0, stream>>>(bufD, Wl_z, bufA, bl_z, NNODES, DIM, 0);
  wmma_gemm_f32<<<gemmGrid, blk, 0, stream>>>(h, Wl_z + DIM * DIM, bufA, nullptr, NNODES, DIM, 1);
  wmma_gemm_f32<<<gemmGrid, blk, 0, stream>>>(bufE, Wl_r, bufB, bl_r, NNODES, DIM, 0);
  wmma_gemm_f32<<<gemmGrid, blk, 0, stream>>>(h, Wl_r + DIM * DIM, bufB, nullptr, NNODES, DIM, 1);
  gates_zr<<<featGrid, blk, 0, stream>>>(bufA, bufB, h, bufC);

  // 5) Hpre = agg_hh @ Wl_hh[:128] + (h*R) @ Wl_hh[128:] + bl_hh
  wmma_gemm_f32<<<gemmGrid, blk, 0, stream>>>(bufF, Wl_h, bufD, bl_h, NNODES, DIM, 0);
  wmma_gemm_f32<<<gemmGrid, blk, 0, stream>>>(bufC, Wl_h + DIM * DIM, bufD, nullptr, NNODES, DIM, 1);

  // 6) h0 = Z*h + (1-Z)*tanh(Hpre); relu(h0) staged for the output linear
  gru_combine<<<featGrid, blk, 0, stream>>>(bufA, bufD, h, out_h0, bufE);

  // 7) z_out = relu(h0) @ W_lin + b_lin
  wmma_gemm_f32<<<gemmGrid, blk, 0, stream>>>(bufE, W_lin, out_z, b_lin, NNODES, DIM, 0);
}